// PSAAttention_31258771980508
// MI455X (gfx1250) — compile-verified
//
#include <hip/hip_runtime.h>
#include <hip/hip_bf16.h>
#include <math.h>

typedef __attribute__((ext_vector_type(2))) float v2f;
typedef __attribute__((ext_vector_type(8))) float v8f;

#define H_DIM 16
#define L_DIM 2048
#define D_DIM 128
#define NQB 16           // 2048/128
#define NKB 32           // 2048/64
#define LDS_STRIDE 130   // 128 + 2 pad to dodge bank conflicts (64 banks)

// ---------------------------------------------------------------------------
// Kernel 1: rank-based top-k selection. rank = #{j: r[j]>r[i]} + ties-by-index.
// Produces the *set* (order irrelevant: pooling sums/means over samples).
// ---------------------------------------------------------------------------
__global__ __launch_bounds__(128)
void psa_topk_kernel(const float* __restrict__ rq, const float* __restrict__ rk,
                     int* __restrict__ selq, int* __restrict__ selk) {
  __shared__ float sq[128];
  __shared__ float sk[64];
  const int h = blockIdx.x, t = threadIdx.x;
  sq[t] = rq[h * 128 + t];
  if (t < 64) sk[t] = rk[h * 64 + t];
  __syncthreads();
  {
    float mv = sq[t]; int rank = 0;
    for (int j = 0; j < 128; ++j)
      rank += (sq[j] > mv) || (sq[j] == mv && j < t);
    if (rank < 32) selq[h * 32 + rank] = t;
  }
  if (t < 64) {
    float mv = sk[t]; int rank = 0;
    for (int j = 0; j < 64; ++j)
      rank += (sk[j] > mv) || (sk[j] == mv && j < t);
    if (rank < 16) selk[h * 16 + rank] = t;
  }
}

// ---------------------------------------------------------------------------
// Kernel 2: pooled sampled attention + level map. One block per (h, q-block).
// 32 sampled queries x 512 sampled keys, softmax over all 512, per-k-block
// mass accumulated, then ranked into levels {0,1,2,8} in-block.
// ---------------------------------------------------------------------------
__global__ __launch_bounds__(256)
void psa_pool_level_kernel(const float* __restrict__ q, const float* __restrict__ k,
                           const int* __restrict__ selq, const int* __restrict__ selk,
                           int* __restrict__ level) {
  __shared__ float qvec[128];
  __shared__ float logits[512];
  __shared__ float red[256];
  __shared__ float sblk[32];
  __shared__ float poolf[32];
  __shared__ int sq_s[32];
  __shared__ int sk_s[16];

  const int h = blockIdx.x >> 4, qb = blockIdx.x & 15, t = threadIdx.x;
  if (t < 32) sq_s[t] = selq[h * 32 + t];
  if (t < 16) sk_s[t] = selk[h * 16 + t];
  __syncthreads();

  const float scale = 0.08838834764831843f;  // 1/sqrt(128)
  const float* kh = k + (size_t)h * L_DIM * D_DIM;
  // This thread owns sampled keys kk = t and kk = t+256.
  const int krow0 = (t >> 4) * 64 + sk_s[t & 15];
  const int krow1 = ((t >> 4) + 16) * 64 + sk_s[t & 15];
  const float* k0 = kh + (size_t)krow0 * D_DIM;
  const float* k1 = kh + (size_t)krow1 * D_DIM;

  float pacc = 0.f;
  for (int qi = 0; qi < 32; ++qi) {
    if (t < 128) qvec[t] = q[((size_t)h * L_DIM + qb * 128 + sq_s[qi]) * D_DIM + t];
    __syncthreads();
    float d0 = 0.f, d1 = 0.f;
    for (int d = 0; d < 128; ++d) { float qd = qvec[d]; d0 += qd * k0[d]; d1 += qd * k1[d]; }
    logits[t] = d0 * scale;
    logits[t + 256] = d1 * scale;
    __syncthreads();
    red[t] = fmaxf(logits[t], logits[t + 256]);
    __syncthreads();
    for (int off = 128; off > 0; off >>= 1) {
      if (t < off) red[t] = fmaxf(red[t], red[t + off]);
      __syncthreads();
    }
    const float mx = red[0];
    logits[t] = __expf(logits[t] - mx);
    logits[t + 256] = __expf(logits[t + 256] - mx);
    __syncthreads();
    float s = 0.f;
    if (t < 32) {
      for (int j = 0; j < 16; ++j) s += logits[t * 16 + j];
      sblk[t] = s;
    }
    __syncthreads();
    if (t < 32) {
      float total = 0.f;
      for (int j = 0; j < 32; ++j) total += sblk[j];
      pacc += s / total;
    }
    __syncthreads();
  }
  if (t < 32) poolf[t] = pacc;   // mean /32 omitted: rank-invariant
  __syncthreads();
  if (t < 32) {
    float mv = poolf[t]; int rank = 0;
    for (int j = 0; j < 32; ++j)
      rank += (poolf[j] > mv) || (poolf[j] == mv && j < t);
    const float frac = (float)rank / 32.0f;
    int lev = 0;
    if (frac < 0.10f)      lev = 1;
    else if (frac < 0.15f) lev = 2;
    else if (frac < 0.35f) lev = 8;   // (1,0,.1)(2,.1,.15)(4,.15,.15)empty(8,.15,.35)
    level[(h * NQB + qb) * NKB + t] = lev;
  }
}

// ---------------------------------------------------------------------------
// Kernel 3: WMMA fp32 flash attention over effective (pooled) keys.
// One WG per (h, q-block); 8 waves, each wave = 16 query rows.
// Repeated pooled logits == +ln(level) bias on distinct pooled keys.
// S^T = K*Q^T so softmax output is directly the A operand of P*V
// (key chunks {2c,2c+1,2c+8,2c+9} -> no cross-lane relayout needed).
// ---------------------------------------------------------------------------
__global__ __launch_bounds__(256)
void psa_attn_kernel(const float* __restrict__ q, const float* __restrict__ k,
                     const float* __restrict__ v, const int* __restrict__ level,
                     float* __restrict__ out) {
  extern __shared__ float smem[];
  float* Qs = smem;                       // [128][130]
  float* Ks = smem + 128 * LDS_STRIDE;    // [16][130]
  float* Vs = Ks + 16 * LDS_STRIDE;       // [16][130]

  const int h = blockIdx.x >> 4, qb = blockIdx.x & 15;
  const int tid = threadIdx.x;
  const int wave = tid >> 5, lane = tid & 31;
  const int hf = lane >> 4, r = lane & 15;

  const float* qh = q + (size_t)h * L_DIM * D_DIM;
  const float* kh = k + (size_t)h * L_DIM * D_DIM;
  const float* vh = v + (size_t)h * L_DIM * D_DIM;

  // Stage the 128x128 fp32 Q tile into LDS (float4 global loads).
  for (int i = tid; i < 128 * 32; i += 256) {
    const int row = i >> 5, dd = (i & 31) << 2;
    const float4 val = *(const float4*)(qh + (size_t)(qb * 128 + row) * D_DIM + dd);
    float* dst = &Qs[row * LDS_STRIDE + dd];
    dst[0] = val.x; dst[1] = val.y; dst[2] = val.z; dst[3] = val.w;
  }
  __syncthreads();

  v8f acc[8];
  const v8f vzero = {0.f, 0.f, 0.f, 0.f, 0.f, 0.f, 0.f, 0.f};
#pragma unroll
  for (int i = 0; i < 8; ++i) acc[i] = vzero;
  float mrun = -__builtin_inff();
  float lrun = 0.f;
  const float scale = 0.08838834764831843f;

  for (int kb = 0; kb < NKB; ++kb) {
    const int lvl = level[(h * NQB + qb) * NKB + kb];
    if (lvl == 0) continue;
    const float bias = (lvl == 1) ? 0.f : ((lvl == 2) ? 0.6931471805599453f
                                                      : 2.0794415416798357f);
    const int nk = 64 / lvl;
    const float invl = 1.f / (float)lvl;

    for (int t0 = 0; t0 < nk; t0 += 16) {
      int nvalid = nk - t0; if (nvalid > 16) nvalid = 16;
      // Cooperative stage of pooled K/V tile (mean over groups of lvl).
      for (int i = tid; i < 2048; i += 256) {
        const int rr = i >> 7, d = i & 127;
        float kvv = 0.f, vvv = 0.f;
        if (rr < nvalid) {
          const size_t base = (size_t)(kb * 64 + (t0 + rr) * lvl) * D_DIM + d;
          for (int u = 0; u < lvl; ++u) { kvv += kh[base + u * D_DIM]; vvv += vh[base + u * D_DIM]; }
          kvv *= invl; vvv *= invl;
        }
        Ks[rr * LDS_STRIDE + d] = kvv;
        Vs[rr * LDS_STRIDE + d] = vvv;
      }
      __syncthreads();

      // S^T (16 keys x 16 queries) = K * Q^T, K-dim = D in chunks of 4.
      v8f c = vzero;
#pragma unroll
      for (int c4 = 0; c4 < 32; ++c4) {
        const int d0 = (c4 << 2) + (hf << 1);
        v2f a = *(const v2f*)&Ks[r * LDS_STRIDE + d0];                  // K row r
        v2f b = *(const v2f*)&Qs[(wave * 16 + r) * LDS_STRIDE + d0];    // Q row
        c = __builtin_amdgcn_wmma_f32_16x16x4_f32(false, a, false, b,
                                                  (short)0, c, false, false);
      }
      // Lane holds S^T[key = j+8*hf][q = wave*16 + r] in c[j].
      float s[8], p[8];
      float tmax = -__builtin_inff();
#pragma unroll
      for (int j = 0; j < 8; ++j) {
        float sv = c[j] * scale + bias;
        if (j + 8 * hf >= nvalid) sv = -1e30f;   // masked padding key
        s[j] = sv;
        tmax = fmaxf(tmax, sv);
      }
      tmax = fmaxf(tmax, __shfl_xor(tmax, 16, 32));
      const float mnew = fmaxf(mrun, tmax);
      const float alpha = __expf(mrun - mnew);
      mrun = mnew;
      float psum = 0.f;
#pragma unroll
      for (int j = 0; j < 8; ++j) { p[j] = __expf(s[j] - mnew); psum += p[j]; }
      psum += __shfl_xor(psum, 16, 32);
      lrun = lrun * alpha + psum;
      // Rescale accumulators: O rows live at row = j+8*hf, alpha lives at lane q=r.
#pragma unroll
      for (int jj = 0; jj < 8; ++jj) {
        const float av = __shfl(alpha, jj + 8 * hf, 32);
#pragma unroll
        for (int dc = 0; dc < 8; ++dc) acc[dc][jj] *= av;
      }
      // P*V: A = P (M=q, K-chunks of keys {2c,2c+1,2c+8,2c+9}), B = V rows.
#pragma unroll
      for (int dc = 0; dc < 8; ++dc) {
        const int dbase = dc * 16 + r;
#pragma unroll
        for (int cc = 0; cc < 4; ++cc) {
          v2f a = {p[2 * cc], p[2 * cc + 1]};
          v2f b = {Vs[(2 * cc + 8 * hf) * LDS_STRIDE + dbase],
                   Vs[(2 * cc + 1 + 8 * hf) * LDS_STRIDE + dbase]};
          acc[dc] = __builtin_amdgcn_wmma_f32_16x16x4_f32(false, a, false, b,
                                                          (short)0, acc[dc], false, false);
        }
      }
      __syncthreads();
    }
  }

  // Normalize and store: lane holds O[row = jj+8*hf][d = dc*16 + r].
  const float linv = 1.f / lrun;
  float* oh = out + ((size_t)h * L_DIM + qb * 128 + wave * 16) * D_DIM;
#pragma unroll
  for (int jj = 0; jj < 8; ++jj) {
    const float li = __shfl(linv, jj + 8 * hf, 32);
    const int row = jj + 8 * hf;
#pragma unroll
    for (int dc = 0; dc < 8; ++dc)
      oh[row * D_DIM + dc * 16 + r] = acc[dc][jj] * li;
  }
}

// ---------------------------------------------------------------------------
extern "C" void kernel_launch(void* const* d_in, const int* in_sizes, int n_in,
                              void* d_out, int out_size, void* d_ws, size_t ws_size,
                              hipStream_t stream) {
  (void)in_sizes; (void)n_in; (void)out_size; (void)ws_size;
  const float* q  = (const float*)d_in[0];
  const float* k  = (const float*)d_in[1];
  const float* v  = (const float*)d_in[2];
  const float* rq = (const float*)d_in[3];
  const float* rk = (const float*)d_in[4];
  float* out = (float*)d_out;

  char* ws = (char*)d_ws;
  int* selq  = (int*)(ws);          // 16*32 ints
  int* selk  = (int*)(ws + 2048);   // 16*16 ints
  int* level = (int*)(ws + 4096);   // 16*16*32 ints  (~36 KB total)

  psa_topk_kernel<<<H_DIM, 128, 0, stream>>>(rq, rk, selq, selk);
  psa_pool_level_kernel<<<H_DIM * NQB, 256, 0, stream>>>(q, k, selq, selk, level);

  const size_t shmem = (size_t)(128 * LDS_STRIDE + 2 * 16 * LDS_STRIDE) * sizeof(float); // 83,200 B
  psa_attn_kernel<<<H_DIM * NQB, 256, shmem, stream>>>(q, k, v, level, out);
}